// GCN_66795331387932
// MI455X (gfx1250) — compile-verified
//
#include <hip/hip_runtime.h>

#define N_NODES 100000
#define N_EDGES 1600000
#define SCAN_BLOCKS ((N_NODES + 255) / 256)   // 391

typedef __attribute__((ext_vector_type(2))) float v2f;
typedef __attribute__((ext_vector_type(4))) float v4f;
typedef __attribute__((ext_vector_type(8))) float v8f;

// ---------------- degree / normalization ----------------

__global__ __launch_bounds__(256) void k_cnt_zero(int* cnt, int n) {
    int i = blockIdx.x * blockDim.x + threadIdx.x;
    if (i < n) cnt[i] = 0;
}

__global__ __launch_bounds__(256) void k_deg_count(const int* __restrict__ ei, int* cnt, int e) {
    int i = blockIdx.x * blockDim.x + threadIdx.x;
    if (i < e) atomicAdd(&cnt[ei[N_EDGES + i]], 1);  // in-degree at dst
}

__global__ __launch_bounds__(256) void k_dinv(const int* __restrict__ cnt, float* dinv, int n) {
    int i = blockIdx.x * blockDim.x + threadIdx.x;
    if (i < n) dinv[i] = rsqrtf((float)cnt[i] + 1.0f);  // +1 self loop; deg>=1 always
}

// ---------------- CSR build: two-level exclusive scan + fill ----------------

__global__ __launch_bounds__(256) void k_chunk_sum(const int* __restrict__ cnt, int* bsum, int n) {
    __shared__ int sm[256];
    int i = blockIdx.x * 256 + threadIdx.x;
    sm[threadIdx.x] = (i < n) ? cnt[i] : 0;
    __syncthreads();
    for (int s = 128; s > 0; s >>= 1) {
        if (threadIdx.x < s) sm[threadIdx.x] += sm[threadIdx.x + s];
        __syncthreads();
    }
    if (threadIdx.x == 0) bsum[blockIdx.x] = sm[0];
}

__global__ void k_scan_blocks(int* bsum, int nb) {
    if (threadIdx.x == 0 && blockIdx.x == 0) {
        int acc = 0;
        for (int b = 0; b < nb; ++b) { int v = bsum[b]; bsum[b] = acc; acc += v; }
    }
}

__global__ __launch_bounds__(256) void k_scan_within(const int* __restrict__ cnt,
                                                     const int* __restrict__ bsum,
                                                     int* rowStart, int* cursor, int n) {
    __shared__ int sm[256];
    int i = blockIdx.x * 256 + threadIdx.x;
    int v = (i < n) ? cnt[i] : 0;
    sm[threadIdx.x] = v;
    __syncthreads();
    for (int ofs = 1; ofs < 256; ofs <<= 1) {        // inclusive Hillis-Steele
        int add = (threadIdx.x >= ofs) ? sm[threadIdx.x - ofs] : 0;
        __syncthreads();
        sm[threadIdx.x] += add;
        __syncthreads();
    }
    if (i < n) {
        int excl = sm[threadIdx.x] - v + bsum[blockIdx.x];
        rowStart[i] = excl;
        cursor[i]   = excl;
    }
}

__global__ __launch_bounds__(256) void k_fill(const int* __restrict__ ei,
                                              const float* __restrict__ dinv,
                                              int* cursor, int* csr_src, float* csr_w, int e) {
    int i = blockIdx.x * blockDim.x + threadIdx.x;
    if (i < e) {
        int s = ei[i], d = ei[N_EDGES + i];
        int pos = atomicAdd(&cursor[d], 1);
        csr_src[pos] = s;
        csr_w[pos]   = dinv[s] * dinv[d];
    }
}

// ---------------- GEMM1: h1 = x @ W1   [N,128]x[128,64]  (fp32 WMMA) ----------------
// block = 128 threads = 4 waves; each block: 16 rows x 64 cols; wave w -> cols [16w,16w+16)

__global__ __launch_bounds__(128) void k_gemm1(const float* __restrict__ x,
                                               const float* __restrict__ W1,
                                               float* __restrict__ h1) {
    __shared__ float xs[16 * 128];   // 8 KB
    __shared__ float ws[128 * 64];   // 32 KB
    const int t = threadIdx.x;
    const int row0 = blockIdx.x * 16;

    {   // stage 16x128 x-tile: 512 float4
        const v4f* s = (const v4f*)(x + row0 * 128);
        v4f* d = (v4f*)xs;
#pragma unroll
        for (int i = 0; i < 4; ++i) d[i * 128 + t] = s[i * 128 + t];
    }
    {   // stage full W1: 2048 float4
        const v4f* s = (const v4f*)W1;
        v4f* d = (v4f*)ws;
#pragma unroll
        for (int i = 0; i < 16; ++i) d[i * 128 + t] = s[i * 128 + t];
    }
    __syncthreads();

    const int lane = t & 31;
    const int wv   = t >> 5;
    const int hl   = lane >> 4;
    const int mn   = lane & 15;
    const int n0   = wv * 16;

    v8f acc = {};
    for (int k = 0; k < 128; k += 4) {
        const int ka = k + hl * 2;
        v2f a, b;
        a.x = xs[mn * 128 + ka];
        a.y = xs[mn * 128 + ka + 1];
        b.x = ws[ka * 64 + n0 + mn];
        b.y = ws[(ka + 1) * 64 + n0 + mn];
        acc = __builtin_amdgcn_wmma_f32_16x16x4_f32(false, a, false, b,
                                                    (short)0, acc, false, false);
    }

    const int mbase = hl * 8;
#pragma unroll
    for (int r = 0; r < 8; ++r)
        h1[(row0 + mbase + r) * 64 + n0 + mn] = acc[r];
}

// ---------------- layer-1 aggregation: gather over CSR (no atomics) ----------------
// one block per node; 64 feature lanes; coalesced 64-float row reads from L2-resident h1

__global__ __launch_bounds__(64) void k_agg1(const float* __restrict__ h1,
                                             const float* __restrict__ dinv,
                                             const int* __restrict__ rowStart,
                                             const int* __restrict__ cnt,
                                             const int* __restrict__ csr_src,
                                             const float* __restrict__ csr_w,
                                             float* __restrict__ agg1) {
    const int i = blockIdx.x;
    const int f = threadIdx.x;
    const float di = dinv[i];
    float acc = h1[i * 64 + f] * di * di;      // self-loop message
    const int beg = rowStart[i];
    const int len = cnt[i];
    for (int j = 0; j < len; ++j) {
        int s = csr_src[beg + j];
        if (j + 1 < len)  // hide dependent-gather latency
            __builtin_prefetch(&h1[csr_src[beg + j + 1] * 64 + f], 0, 1);
        acc = fmaf(h1[s * 64 + f], csr_w[beg + j], acc);
    }
    agg1[i * 64 + f] = acc;
}

// ---------------- GEMM2: h3 = relu(agg1+b1) @ W2   [N,64]x[64,40]  (fp32 WMMA) ----------------

__global__ __launch_bounds__(96) void k_gemm2(const float* __restrict__ agg1,
                                              const float* __restrict__ b1,
                                              const float* __restrict__ W2,
                                              float* __restrict__ h3) {
    __shared__ float as[16 * 64];
    __shared__ float ws[64 * 48];
    const int t = threadIdx.x;
    const int row0 = blockIdx.x * 16;

    for (int idx = t; idx < 16 * 64; idx += 96) {
        int col = idx & 63;
        float v = agg1[row0 * 64 + idx] + b1[col];
        as[idx] = v > 0.0f ? v : 0.0f;          // fused bias + relu
    }
    for (int idx = t; idx < 64 * 48; idx += 96) {
        int k = idx / 48, n = idx - k * 48;
        ws[idx] = (n < 40) ? W2[k * 40 + n] : 0.0f;
    }
    __syncthreads();

    const int lane = t & 31;
    const int wv   = t >> 5;
    const int hl   = lane >> 4;
    const int mn   = lane & 15;
    const int n0   = wv * 16;

    v8f acc = {};
    for (int k = 0; k < 64; k += 4) {
        const int ka = k + hl * 2;
        v2f a, b;
        a.x = as[mn * 64 + ka];
        a.y = as[mn * 64 + ka + 1];
        b.x = ws[ka * 48 + n0 + mn];
        b.y = ws[(ka + 1) * 48 + n0 + mn];
        acc = __builtin_amdgcn_wmma_f32_16x16x4_f32(false, a, false, b,
                                                    (short)0, acc, false, false);
    }

    const int col = n0 + mn;
    const int mbase = hl * 8;
    if (col < 40) {
#pragma unroll
        for (int r = 0; r < 8; ++r)
            h3[(row0 + mbase + r) * 40 + col] = acc[r];
    }
}

// ---------------- layer-2 aggregation: gather over CSR into d_out ----------------

__global__ __launch_bounds__(64) void k_agg2(const float* __restrict__ h3,
                                             const float* __restrict__ dinv,
                                             const float* __restrict__ b2,
                                             const int* __restrict__ rowStart,
                                             const int* __restrict__ cnt,
                                             const int* __restrict__ csr_src,
                                             const float* __restrict__ csr_w,
                                             float* __restrict__ out) {
    const int i = blockIdx.x;
    const int f = threadIdx.x;
    if (f >= 40) return;
    const float di = dinv[i];
    float acc = b2[f] + h3[i * 40 + f] * di * di;   // bias + self loop
    const int beg = rowStart[i];
    const int len = cnt[i];
    for (int j = 0; j < len; ++j) {
        int s = csr_src[beg + j];
        if (j + 1 < len)
            __builtin_prefetch(&h3[csr_src[beg + j + 1] * 40 + f], 0, 1);
        acc = fmaf(h3[s * 40 + f], csr_w[beg + j], acc);
    }
    out[i * 40 + f] = acc;
}

// ---------------- launcher ----------------

extern "C" void kernel_launch(void* const* d_in, const int* in_sizes, int n_in,
                              void* d_out, int out_size, void* d_ws, size_t ws_size,
                              hipStream_t stream) {
    const float* x  = (const float*)d_in[0];
    const int*   ei = (const int*)d_in[1];   // [2, E] row-major: src then dst
    const float* W1 = (const float*)d_in[2];
    const float* b1 = (const float*)d_in[3];
    const float* W2 = (const float*)d_in[4];
    const float* b2 = (const float*)d_in[5];
    float* out = (float*)d_out;

    // workspace layout (4-byte units, all 16B-aligned)
    float* wsf      = (float*)d_ws;
    float* dinv     = wsf;                                   // N
    int*   cnt      = (int*)(wsf + N_NODES);                 // N
    int*   rowStart = cnt + N_NODES;                         // N
    int*   cursor   = rowStart + N_NODES;                    // N
    int*   bsum     = cursor + N_NODES;                      // 512 (pad)
    int*   csr_src  = bsum + 512;                            // E
    float* csr_w    = (float*)(csr_src + N_EDGES);           // E
    float* h1       = csr_w + N_EDGES;                       // N*64
    float* agg1     = h1 + (size_t)N_NODES * 64;             // N*64
    float* h3       = agg1 + (size_t)N_NODES * 64;           // N*40

    // 1) degrees + normalization
    k_cnt_zero   <<<(N_NODES + 255) / 256, 256, 0, stream>>>(cnt, N_NODES);
    k_deg_count  <<<(N_EDGES + 255) / 256, 256, 0, stream>>>(ei, cnt, N_EDGES);
    k_dinv       <<<(N_NODES + 255) / 256, 256, 0, stream>>>(cnt, dinv, N_NODES);

    // 2) CSR by destination
    k_chunk_sum  <<<SCAN_BLOCKS, 256, 0, stream>>>(cnt, bsum, N_NODES);
    k_scan_blocks<<<1, 32, 0, stream>>>(bsum, SCAN_BLOCKS);
    k_scan_within<<<SCAN_BLOCKS, 256, 0, stream>>>(cnt, bsum, rowStart, cursor, N_NODES);
    k_fill       <<<(N_EDGES + 255) / 256, 256, 0, stream>>>(ei, dinv, cursor, csr_src, csr_w, N_EDGES);

    // 3) layer 1: WMMA GEMM + gather aggregation
    k_gemm1      <<<N_NODES / 16, 128, 0, stream>>>(x, W1, h1);
    k_agg1       <<<N_NODES, 64, 0, stream>>>(h1, dinv, rowStart, cnt, csr_src, csr_w, agg1);

    // 4) layer 2: WMMA GEMM (fused relu+bias) + gather aggregation into out
    k_gemm2      <<<N_NODES / 16, 96, 0, stream>>>(agg1, b1, W2, h3);
    k_agg2       <<<N_NODES, 64, 0, stream>>>(h3, dinv, b2, rowStart, cnt, csr_src, csr_w, out);
}